// BPR_LOSS_11347303596571
// MI455X (gfx1250) — compile-verified
//
#include <hip/hip_runtime.h>

#define MAXLEN 256
#define NS 3          // soft labels s = 1..3
#define NWAVES 8      // 256 threads / wave32

typedef __attribute__((ext_vector_type(2))) float v2f;
typedef __attribute__((ext_vector_type(8))) float v8f;

// log(sigmoid(x)) = min(x,0) - ln2 * log2(1 + exp2(-|x|*log2e))
// Raw v_exp_f32 / v_log_f32: argument of log2 is in (1,2] -> no denorm/inf
// guards needed, so we bypass the OCML range-check code entirely.
__device__ __forceinline__ float logsig(float x) {
    const float t = __builtin_amdgcn_exp2f(fabsf(x) * -1.44269504088896340f);
    return fminf(x, 0.0f)
         - 0.69314718055994531f * __builtin_amdgcn_logf(1.0f + t);
}

__global__ __launch_bounds__(256) void bpr_graph_kernel(
    const float* __restrict__ logits,
    const int*   __restrict__ labels,
    const int*   __restrict__ s_num,
    float* __restrict__ graph_lp,
    float* __restrict__ graph_valid)
{
    __shared__ float s_lg[MAXLEN];              // padded logits
    __shared__ float s_posz[NS + 1][MAXLEN];    // pos weights; row NS = zeros
    __shared__ float s_negz[NS + 1][MAXLEN];    // neg weights; row NS = zeros
    __shared__ int   s_red[256];                // offset reduction scratch
    __shared__ int   s_npos[NS];
    __shared__ int   s_nneg[NS];
    __shared__ float s_wps[NWAVES][2 * NS];     // per-wave partial pair sums

    const int b = blockIdx.x;
    const int t = threadIdx.x;

    // ---- segment offset = sum_{i<b} s_num[i] (block tree reduction) ----
    int part = 0;
    for (int i = t; i < b; i += 256) part += s_num[i];
    s_red[t] = part;
    __syncthreads();
    for (int o = 128; o > 0; o >>= 1) {
        if (t < o) s_red[t] += s_red[t + o];
        __syncthreads();
    }
    const int offset = s_red[0];
    int L = s_num[b];
    if (L > MAXLEN) L = MAXLEN;

    // ---- init counters / partials ----
    if (t < NS) { s_npos[t] = 0; s_nneg[t] = 0; }
    if (t < NWAVES * 2 * NS) ((float*)s_wps)[t] = 0.0f;

    // ---- stage logits / label-derived weights into LDS (padded with 0) ----
    s_posz[NS][t] = 0.0f;                       // zero row for lanes with col>=NS
    s_negz[NS][t] = 0.0f;
    if (t < L) {
        const float lv  = logits[offset + t];
        const int   lbv = labels[offset + t];
        s_lg[t] = lv;
#pragma unroll
        for (int s = 0; s < NS; ++s) {
            const float pw = (lbv == s + 1) ? 1.0f : 0.0f;
            const float nw = (lbv <  s + 1) ? 1.0f : 0.0f;
            s_posz[s][t] = pw;
            s_negz[s][t] = nw;
            if (pw != 0.0f) atomicAdd(&s_npos[s], 1);
            if (nw != 0.0f) atomicAdd(&s_nneg[s], 1);
        }
    } else {
        s_lg[t] = 0.0f;
#pragma unroll
        for (int s = 0; s < NS; ++s) { s_posz[s][t] = 0.0f; s_negz[s][t] = 0.0f; }
    }
    __syncthreads();

    // ---- pairwise bilinear reduction via V_WMMA_F32_16X16X4_F32 ----
    // D[m][s] += sum_k LS[p_tile+m][4c+k] * neg_{s+1}[4c+k]
    const int lane = t & 31;
    const int wave = t >> 5;
    const int pl   = lane & 15;                 // A: M row / B,D: N column
    const int half = lane >> 4;                 // 0 or 1
    const int kh   = half << 1;                 // A/B K sub-rows: {0,1} or {2,3}
    const int cz   = (pl < NS) ? pl : NS;       // column -> weight row (NS = zeros)

    const int ptiles  = (L + 15) >> 4;
    const int nchunks = (L + 3) >> 2;           // max n index = 4*nchunks-1 <= 255

    float wacc = 0.0f;                          // lane partial of pos^T * D col
    for (int pi = wave; pi < ptiles; pi += NWAVES) {
        const int   p   = pi * 16 + pl;
        const float lgp = s_lg[p];
        v8f acc = {0.f, 0.f, 0.f, 0.f, 0.f, 0.f, 0.f, 0.f};
        for (int c = 0; c < nchunks; ++c) {
            const int n0 = (c << 2) + kh;       // even -> 8B-aligned pair loads
            const int n1 = n0 + 1;
            v2f a, bb;
            a.x  = logsig(lgp - s_lg[n0]);
            a.y  = logsig(lgp - s_lg[n1]);
            bb.x = s_negz[cz][n0];
            bb.y = s_negz[cz][n1];
            acc = __builtin_amdgcn_wmma_f32_16x16x4_f32(
                false, a, false, bb, (short)0, acc, false, false);
        }
        // tile epilogue: tsum_s = sum_m pos_s[m] * D[m][s] for this lane's half
        const int mbase = pi * 16 + (half << 3);
        float tsum = 0.0f;
#pragma unroll
        for (int r = 0; r < 8; ++r)
            tsum += s_posz[cz][mbase + r] * acc[r];
        wacc += tsum;
    }
    if (pl < NS) s_wps[wave][pl + NS * half] = wacc;
    __syncthreads();

    // ---- per-graph epilogue (fixed summation order => deterministic) ----
    if (t == 0) {
        float means_sum = 0.0f;
        int   cnt = 0;
#pragma unroll
        for (int s = 0; s < NS; ++s) {
            float ps = 0.0f;
            for (int w = 0; w < NWAVES; ++w)
                ps += s_wps[w][s] + s_wps[w][s + NS];
            const int np = s_npos[s];
            const int nn = s_nneg[s];
            if (np > 0 && nn > 0) {
                means_sum += ps / ((float)np * (float)nn);
                ++cnt;
            }
        }
        const float lp = (cnt > 0) ? means_sum / (float)cnt : 0.0f;
        graph_lp[b]    = lp;
        graph_valid[b] = (cnt > 0) ? 1.0f : 0.0f;
    }
}

__global__ __launch_bounds__(512) void bpr_final_kernel(
    const float* __restrict__ graph_lp,
    const float* __restrict__ graph_valid,
    int nB,
    float* __restrict__ out)
{
    __shared__ float r1[512];
    __shared__ float r2[512];
    const int t = threadIdx.x;
    float a = 0.0f, v = 0.0f;
    for (int i = t; i < nB; i += 512) { a += graph_lp[i]; v += graph_valid[i]; }
    r1[t] = a; r2[t] = v;
    __syncthreads();
    for (int o = 256; o > 0; o >>= 1) {
        if (t < o) { r1[t] += r1[t + o]; r2[t] += r2[t + o]; }
        __syncthreads();
    }
    if (t == 0) {
        const float nv = (r2[0] > 1.0f) ? r2[0] : 1.0f;
        out[0] = -(r1[0] / nv);
    }
}

extern "C" void kernel_launch(void* const* d_in, const int* in_sizes, int n_in,
                              void* d_out, int out_size, void* d_ws, size_t ws_size,
                              hipStream_t stream) {
    const float* logits = (const float*)d_in[0];
    const int*   labels = (const int*)d_in[1];
    const int*   s_num  = (const int*)d_in[2];
    const int nB = in_sizes[2];

    float* graph_lp    = (float*)d_ws;
    float* graph_valid = graph_lp + nB;

    bpr_graph_kernel<<<nB, 256, 0, stream>>>(logits, labels, s_num,
                                             graph_lp, graph_valid);
    bpr_final_kernel<<<1, 512, 0, stream>>>(graph_lp, graph_valid, nB,
                                            (float*)d_out);
}